// HPSS_46136538693669
// MI455X (gfx1250) — compile-verified
//
#include <hip/hip_runtime.h>

// HPSS (harmonic/percussive source separation), fused single kernel.
// S: (4 planes, H=1025, W=1024) f32. Two 31-tap median filters + soft masks.
// Compute-bound on the median selection -> forgetful selection (~255 CE/median).
// Staging: Tensor Data Mover (tensor_load_to_lds) for interior tiles,
// CDNA5 async global->LDS b32 copies for left/top edge tiles.

#define TILE     64
#define HALO     15                 // (31-1)/2
#define REG      (TILE + 2 * HALO)  // 94 = LDS tile extent and row pitch
#define NTHREADS 256

#if __has_builtin(__builtin_amdgcn_global_load_async_to_lds_b32)
#define HAVE_ASYNC_LDS 1
#else
#define HAVE_ASYNC_LDS 0
#endif

#if __has_builtin(__builtin_amdgcn_tensor_load_to_lds) && \
    __has_builtin(__builtin_amdgcn_s_wait_tensorcnt)
#define HAVE_TDM 1
#else
#define HAVE_TDM 0
#endif

typedef __attribute__((address_space(1))) int* gptr_i32;
typedef __attribute__((address_space(3))) int* lptr_i32;
typedef unsigned int u32x4 __attribute__((ext_vector_type(4)));
typedef int i32x4 __attribute__((ext_vector_type(4)));
typedef int i32x8 __attribute__((ext_vector_type(8)));

__device__ __forceinline__ void cswap(float& a, float& b) {
    float lo = fminf(a, b);
    float hi = fmaxf(a, b);
    a = lo;
    b = hi;
}

// Median of 31 values read from LDS at compile-time stride.
// Forgetful selection: buffer of 17; each round drops running min & max and
// inserts the next element; the middle of the final 3 is the median.
template <int STRIDE>
__device__ __forceinline__ float median31(const float* p) {
    float b[17];
#pragma unroll
    for (int i = 0; i < 17; ++i) b[i] = p[i * STRIDE];
#pragma unroll
    for (int n = 17; n < 31; ++n) {
        const int s = 34 - n;  // effective buffer size: 17 down to 4
#pragma unroll
        for (int i = 0; i < s - 1; ++i) cswap(b[i], b[i + 1]);  // max -> b[s-1]
#pragma unroll
        for (int i = s - 2; i > 0; --i) cswap(b[i - 1], b[i]);  // min -> b[0]
        b[0] = p[n * STRIDE];  // overwrite min; max dropped by shrinking
    }
    cswap(b[0], b[1]);
    cswap(b[1], b[2]);
    cswap(b[0], b[1]);
    return b[1];  // middle of the 3 survivors
}

__global__ __launch_bounds__(NTHREADS) void hpss_fused(
    const float* __restrict__ S, float* __restrict__ outH,
    float* __restrict__ outP, int H, int W) {
    __shared__ float tile[REG * REG];

    const int t   = (int)threadIdx.x;
    const int gx0 = (int)blockIdx.x * TILE - HALO;
    const int gy0 = (int)blockIdx.y * TILE - HALO;
    const size_t plane = (size_t)H * (size_t)W;
    const float* Sp = S + (size_t)blockIdx.z * plane;
    float* Hp = outH + (size_t)blockIdx.z * plane;
    float* Pp = outP + (size_t)blockIdx.z * plane;

#if HAVE_TDM
    const bool use_tdm = (gx0 >= 0) && (gy0 >= 0);
    if (use_tdm) {
        // One TDM descriptor per block: 94x94 f32 tile, row stride W.
        // Right/bottom zero padding comes free: reads past tensor_dim return 0
        // while the LDS write address keeps advancing (pitch preserved).
        if (t < 32) {  // wave 0 issues the DMA
            const unsigned lds_base =
                (unsigned)(uintptr_t)(lptr_i32)&tile[0];
            const unsigned long long ga =
                (unsigned long long)(uintptr_t)(Sp + (size_t)gy0 * W + gx0);
            const unsigned tdim0 = (unsigned)(W - gx0);  // valid extent in x
            const unsigned tdim1 = (unsigned)(H - gy0);  // valid extent in y

            u32x4 g0;
            g0[0] = 1u;                         // count=1, user mode
            g0[1] = lds_base;                   // lds_addr
            g0[2] = (unsigned)ga;               // global_addr[31:0]
            g0[3] = (unsigned)((ga >> 32) & 0x01FFFFFFu) | (2u << 30);  // type=2

            i32x8 g1;
            g1[0] = (int)(2u << 16);            // data_size=2 (4 bytes)
            g1[1] = (int)((tdim0 & 0xFFFFu) << 16);          // tensor_dim0 lo
            g1[2] = (int)((tdim0 >> 16) | ((tdim1 & 0xFFFFu) << 16));
            g1[3] = (int)((tdim1 >> 16) | ((unsigned)REG << 16));  // tile_dim0
            g1[4] = (int)REG;                   // tile_dim1 (tile_dim2=0)
            g1[5] = (int)(unsigned)W;           // tensor_dim0_stride lo32
            g1[6] = 0;                          // stride0 hi16 | stride1 lo16
            g1[7] = 0;                          // stride1 hi32

            i32x4 gz = {0, 0, 0, 0};
#if defined(__clang_major__) && __clang_major__ >= 23
            i32x8 gz8 = {0, 0, 0, 0, 0, 0, 0, 0};
            __builtin_amdgcn_tensor_load_to_lds(g0, g1, gz, gz, gz8, 0);
#else
            __builtin_amdgcn_tensor_load_to_lds(g0, g1, gz, gz, 0);
#endif
            __builtin_amdgcn_s_wait_tensorcnt(0);
        }
    } else
#endif  // HAVE_TDM
    {
        // Edge tiles (negative origin): pre-zero (zero padding), then stage
        // the valid region with async global->LDS copies.
#pragma unroll 4
        for (int i = t; i < REG * REG; i += NTHREADS) tile[i] = 0.0f;
        __syncthreads();

        for (int e = t; e < REG * REG; e += NTHREADS) {
            const int r  = e / REG;
            const int c  = e - r * REG;
            const int gy = gy0 + r;
            const int gx = gx0 + c;
            if ((unsigned)gy < (unsigned)H && (unsigned)gx < (unsigned)W) {
#if HAVE_ASYNC_LDS
                __builtin_amdgcn_global_load_async_to_lds_b32(
                    (gptr_i32)(Sp + (size_t)gy * W + gx),
                    (lptr_i32)&tile[r * REG + c],
                    /*offset=*/0, /*cpol=*/0);
#else
                tile[r * REG + c] = Sp[(size_t)gy * W + gx];
#endif
            }
        }
#if HAVE_ASYNC_LDS
#if __has_builtin(__builtin_amdgcn_s_wait_asynccnt)
        __builtin_amdgcn_s_wait_asynccnt(0);
#else
        asm volatile("s_wait_asynccnt 0" ::: "memory");
#endif
#endif
    }
    __syncthreads();

    // Each thread: 16 outputs in one column of the 64x64 tile.
    const int lx  = t & (TILE - 1);
    const int ly0 = (t >> 6) * (TILE / 4);
    const int gx  = (int)blockIdx.x * TILE + lx;
    if (gx >= W) return;

#pragma unroll 1
    for (int i = 0; i < TILE / 4; ++i) {
        const int ly = ly0 + i;
        const int gy = (int)blockIdx.y * TILE + ly;
        if (gy >= H) continue;

        const float* rowp = &tile[(ly + HALO) * REG + lx];   // along time (W)
        const float* colp = &tile[ly * REG + (lx + HALO)];   // along freq (H)
        const float h = median31<1>(rowp);
        const float p = median31<REG>(colp);
        const float s = tile[(ly + HALO) * REG + (lx + HALO)];

        // softmask, power=2, margin=1 (mirrors reference algebra exactly)
        float Z = fmaxf(h, p);
        if (Z < 1.17549435e-38f) Z = 1.0f;  // finfo(f32).tiny
        const float hz = h / Z, pz = p / Z;
        const float m = hz * hz, r = pz * pz;
        const float inv = 1.0f / (m + r);

        const size_t o = (size_t)gy * (size_t)W + (size_t)gx;
        __builtin_nontemporal_store(s * (m * inv), &Hp[o]);
        __builtin_nontemporal_store(s * (r * inv), &Pp[o]);
    }
}

extern "C" void kernel_launch(void* const* d_in, const int* in_sizes, int n_in,
                              void* d_out, int out_size, void* d_ws, size_t ws_size,
                              hipStream_t stream) {
    (void)in_sizes; (void)n_in; (void)d_ws; (void)ws_size; (void)out_size;
    const float* S = (const float*)d_in[0];
    float* out = (float*)d_out;

    const int B = 2, C = 2, H = 1025, W = 1024;
    const size_t per = (size_t)B * C * H * W;  // 4,198,400 elements per output
    float* outH = out;        // S * mask_harm
    float* outP = out + per;  // S * mask_perc

    dim3 grid((W + TILE - 1) / TILE, (H + TILE - 1) / TILE, B * C);
    hpss_fused<<<grid, NTHREADS, 0, stream>>>(S, outH, outP, H, W);
}